// SimpleHGAT_64836826301214
// MI455X (gfx1250) — compile-verified
//
#include <hip/hip_runtime.h>
#include <cstdint>
#include <cstddef>

// ---------------------------------------------------------------------------
// SimpleHGAT for MI455X (gfx1250). fp32 end-to-end (matches reference dtype).
// Matrix work on V_WMMA_F32_16X16X4_F32; everything else is bandwidth-bound
// gather/scatter with f32/u32 atomics.
// ---------------------------------------------------------------------------

typedef __attribute__((ext_vector_type(2))) float v2f;
typedef __attribute__((ext_vector_type(8))) float v8f;

#define NN 100000        // nodes
#define NE 1600000       // edges
#define HD 64            // H*D = H*DE = DIN
#define NH 4             // heads

// ---- monotone float <-> uint encoding (order-preserving) ----
__device__ __forceinline__ unsigned enc_f32(float f) {
    unsigned b = __float_as_uint(f);
    return (b & 0x80000000u) ? ~b : (b | 0x80000000u);
}
__device__ __forceinline__ float dec_f32(unsigned e) {
    unsigned b = (e & 0x80000000u) ? (e & 0x7FFFFFFFu) : ~e;
    return __uint_as_float(b);
}

// ---------------------------------------------------------------------------
__global__ void k_init_node_stats(unsigned* __restrict__ segmax,
                                  float* __restrict__ denom) {
    int t = blockIdx.x * blockDim.x + threadIdx.x;
    if (t < NN * NH) { segmax[t] = 0u; denom[t] = 0.0f; }
}

__global__ void k_init_out(const float* __restrict__ xn,
                           const float* __restrict__ bias,
                           float* __restrict__ out) {
    int t = blockIdx.x * blockDim.x + threadIdx.x;
    if (t < NN * HD) out[t] = xn[t] + bias[t & (HD - 1)];
}

// V[k][h] = sum_d W_edge[k, h*16+d] * attn_edge[h, d]   (64x4)
__global__ void k_fold_edge_attn(const float* __restrict__ W_edge,
                                 const float* __restrict__ attn_edge,
                                 float* __restrict__ V) {
    int t = threadIdx.x;             // 256 threads, one block
    int k = t >> 2, h = t & 3;
    float s = 0.0f;
#pragma unroll
    for (int d = 0; d < 16; ++d)
        s += W_edge[k * HD + h * 16 + d] * attn_edge[h * 16 + d];
    V[k * NH + h] = s;
}

// ---------------------------------------------------------------------------
// hn = X (100000x64) @ W (64x64), one wave per 16-row tile.
// A (16x4 f32): lanes 0-15 M=0-15; vgpr0 = K={0|2}, vgpr1 = K={1|3} by lane half.
// B (4x16 f32): lane n = l&15; vgpr0 = row {0|2}, vgpr1 = row {1|3}.
// C (16x16 f32): vgpr v: rows v (lanes 0-15) / v+8 (lanes 16-31), col = l&15.
// ---------------------------------------------------------------------------
__global__ void __launch_bounds__(256)
k_node_gemm(const float* __restrict__ X, const float* __restrict__ W,
            float* __restrict__ HN) {
    const int lane = threadIdx.x & 31;
    const int wave = threadIdx.x >> 5;
    const int tile = blockIdx.x * 8 + wave;           // 6250 tiles
    if (tile >= NN / 16) return;                      // wave-uniform exit
    const int m  = lane & 15;
    const int kh = lane >> 4;
    const int n  = lane & 15;

    const float* xrow = X + (size_t)(tile * 16 + m) * HD;
    v2f a[16];
#pragma unroll
    for (int kk = 0; kk < 16; ++kk) {
        a[kk].x = xrow[kk * 4 + 2 * kh];
        a[kk].y = xrow[kk * 4 + 2 * kh + 1];
    }
#pragma unroll
    for (int nt = 0; nt < 4; ++nt) {                  // 4 N-tiles == 4 heads
        v8f c = {};
#pragma unroll
        for (int kk = 0; kk < 16; ++kk) {
            v2f b;
            b.x = W[(kk * 4 + 2 * kh)     * HD + nt * 16 + n];
            b.y = W[(kk * 4 + 2 * kh + 1) * HD + nt * 16 + n];
            c = __builtin_amdgcn_wmma_f32_16x16x4_f32(
                    false, a[kk], false, b, (short)0, c, false, false);
        }
#pragma unroll
        for (int v = 0; v < 8; ++v) {
            int row = v + 8 * kh;
            HN[(size_t)(tile * 16 + row) * HD + nt * 16 + n] = c[v];
        }
    }
}

// w_e = X_e (1.6Mx64) @ V (64x4), B padded to 16 cols with zeros.
__global__ void __launch_bounds__(256)
k_edge_gemm(const float* __restrict__ XE, const float* __restrict__ V,
            float* __restrict__ WE) {
    const int lane = threadIdx.x & 31;
    const int wave = threadIdx.x >> 5;
    const int tile = blockIdx.x * 8 + wave;           // 100000 tiles (exact)
    const int m  = lane & 15;
    const int kh = lane >> 4;
    const int n  = lane & 15;

    const float* xrow = XE + (size_t)(tile * 16 + m) * HD;
    v8f c = {};
#pragma unroll
    for (int kk = 0; kk < 16; ++kk) {
        v2f a, b;
        a.x = xrow[kk * 4 + 2 * kh];
        a.y = xrow[kk * 4 + 2 * kh + 1];
        b.x = (n < NH) ? V[(kk * 4 + 2 * kh)     * NH + n] : 0.0f;
        b.y = (n < NH) ? V[(kk * 4 + 2 * kh + 1) * NH + n] : 0.0f;
        c = __builtin_amdgcn_wmma_f32_16x16x4_f32(
                false, a, false, b, (short)0, c, false, false);
    }
    if (n < NH) {
#pragma unroll
        for (int v = 0; v < 8; ++v) {
            int row = v + 8 * kh;
            WE[(size_t)(tile * 16 + row) * NH + n] = c[v];
        }
    }
}

// w_q[n,h] = <hn[n,h,:], attn_query[h,:]>, same for w_k.
__global__ void k_wqwk(const float* __restrict__ HN,
                       const float* __restrict__ aq,
                       const float* __restrict__ ak,
                       float* __restrict__ wq, float* __restrict__ wk) {
    int t = blockIdx.x * blockDim.x + threadIdx.x;
    if (t >= NN * NH) return;
    int node = t >> 2, h = t & 3;
    const float* hrow = HN + (size_t)node * HD + h * 16;
    float sq = 0.0f, sk = 0.0f;
#pragma unroll
    for (int d = 0; d < 16; ++d) {
        float x = hrow[d];
        sq += x * aq[h * 16 + d];
        sk += x * ak[h * 16 + d];
    }
    wq[t] = sq;
    wk[t] = sk;
}

// logits = leaky_relu(wk[src] + wq[dst] + we), fused segment-max (encoded u32).
__global__ void k_logits(const int* __restrict__ src, const int* __restrict__ dst,
                         const float* __restrict__ wq, const float* __restrict__ wk,
                         const float* __restrict__ we, float* __restrict__ logits,
                         unsigned* __restrict__ segmax) {
    int e = blockIdx.x * blockDim.x + threadIdx.x;
    if (e >= NE) return;
    int s = src[e], d = dst[e];
#pragma unroll
    for (int h = 0; h < NH; ++h) {
        float x = wk[s * NH + h] + wq[d * NH + h] + we[e * NH + h];
        x = (x >= 0.0f) ? x : 0.01f * x;
        logits[e * NH + h] = x;
        atomicMax(&segmax[d * NH + h], enc_f32(x));
    }
}

// ex = exp(logit - segmax[dst]) in place, denom += ex.
__global__ void k_exp_denom(const int* __restrict__ dst,
                            const unsigned* __restrict__ segmax,
                            float* __restrict__ logits,
                            float* __restrict__ denom) {
    int t = blockIdx.x * blockDim.x + threadIdx.x;
    if (t >= NE * NH) return;
    int e = t >> 2, h = t & 3;
    int d = dst[e];
    float m  = dec_f32(segmax[d * NH + h]);
    float ex = __expf(logits[t] - m);
    logits[t] = ex;
    atomicAdd(&denom[d * NH + h], ex);
}

// out[dst, h, :] += hn[src, h, :] * (ex / denom[dst,h])
__global__ void k_scatter(const int* __restrict__ src, const int* __restrict__ dst,
                          const float* __restrict__ ex,
                          const float* __restrict__ denom,
                          const float* __restrict__ HN,
                          float* __restrict__ out) {
    int t = blockIdx.x * blockDim.x + threadIdx.x;
    if (t >= NE * NH) return;
    int e = t >> 2, h = t & 3;
    int s = src[e], d = dst[e];
    float attn = ex[t] / denom[d * NH + h];
    const float* hrow = HN + (size_t)s * HD + h * 16;
    float* orow = out + (size_t)d * HD + h * 16;
#pragma unroll
    for (int i = 0; i < 16; ++i)
        atomicAdd(&orow[i], hrow[i] * attn);
}

// ---------------------------------------------------------------------------
extern "C" void kernel_launch(void* const* d_in, const int* in_sizes, int n_in,
                              void* d_out, int out_size, void* d_ws, size_t ws_size,
                              hipStream_t stream) {
    const float* inputs_node = (const float*)d_in[0];
    const float* inputs_edge = (const float*)d_in[1];
    const int*   src         = (const int*)  d_in[2];
    const int*   dst         = (const int*)  d_in[3];
    const float* W_node      = (const float*)d_in[4];
    const float* W_edge      = (const float*)d_in[5];
    const float* attn_query  = (const float*)d_in[6];
    const float* attn_key    = (const float*)d_in[7];
    const float* attn_edge   = (const float*)d_in[8];
    const float* bias        = (const float*)d_in[9];
    float* out = (float*)d_out;

    // workspace layout (float units)
    float* ws = (float*)d_ws;
    float*    HN     = ws;                                   // 6,400,000
    float*    WQ     = HN  + (size_t)NN * HD;                //   400,000
    float*    WK     = WQ  + (size_t)NN * NH;                //   400,000
    float*    WE     = WK  + (size_t)NN * NH;                // 6,400,000
    float*    LOGITS = WE  + (size_t)NE * NH;                // 6,400,000 (reused as ex)
    unsigned* SEGMAX = (unsigned*)(LOGITS + (size_t)NE * NH);//   400,000 (u32)
    float*    DENOM  = (float*)(SEGMAX + (size_t)NN * NH);   //   400,000
    float*    V      = DENOM + (size_t)NN * NH;              //       256

    const int TPB = 256;

    // 1) init
    k_init_node_stats<<<(NN * NH + TPB - 1) / TPB, TPB, 0, stream>>>(SEGMAX, DENOM);
    k_init_out<<<(NN * HD + TPB - 1) / TPB, TPB, 0, stream>>>(inputs_node, bias, out);
    k_fold_edge_attn<<<1, 256, 0, stream>>>(W_edge, attn_edge, V);

    // 2) node GEMM (WMMA f32): 6250 tiles, 8 waves/block
    k_node_gemm<<<(NN / 16 + 7) / 8, TPB, 0, stream>>>(inputs_node, W_node, HN);

    // 3) per-node attention scalars
    k_wqwk<<<(NN * NH + TPB - 1) / TPB, TPB, 0, stream>>>(HN, attn_query, attn_key, WQ, WK);

    // 4) edge GEMM (WMMA f32): 100000 tiles, 8 waves/block (exact)
    k_edge_gemm<<<(NE / 16) / 8, TPB, 0, stream>>>(inputs_edge, V, WE);

    // 5) logits + segment max
    k_logits<<<(NE + TPB - 1) / TPB, TPB, 0, stream>>>(src, dst, WQ, WK, WE, LOGITS, SEGMAX);

    // 6) exp + denom
    k_exp_denom<<<(NE * NH + TPB - 1) / TPB, TPB, 0, stream>>>(dst, SEGMAX, LOGITS, DENOM);

    // 7) weighted scatter-sum
    k_scatter<<<(NE * NH + TPB - 1) / TPB, TPB, 0, stream>>>(src, dst, LOGITS, DENOM, HN, out);
}